// EnhancedChannelFilter_manual_17952963297635
// MI455X (gfx1250) — compile-verified
//
#include <hip/hip_runtime.h>
#include <hip/hip_bf16.h>

// ---------------------------------------------------------------------------
// EnhancedChannelFilter — MI455X (gfx1250) fused pipeline
//   K0 : packet-loss mask + channel-mean reduction
//   K0b: SE-MLP -> per-(b,c) gating mask
//   G1 : det = Dw @ xm, epi: xs = sigmoid(det)*xm    (WMMA f32, TDM-staged B)
//   G2 : r1 = relu(R1 @ [xm; xs])                    (WMMA f32, 2-phase TDM)
//   G3 : out = (R2 @ r1) * mask                      (WMMA f32, TDM-staged B)
// B-panels (256 x 64 f32 = 64KB) are DMA'd to LDS by the Tensor Data Mover
// (tensor_load_to_lds, TENSORcnt), WMMA reads B from LDS, A streams from L2.
// ---------------------------------------------------------------------------

typedef __attribute__((ext_vector_type(2))) float v2f;
typedef __attribute__((ext_vector_type(8))) float v8f;
typedef __attribute__((ext_vector_type(4))) unsigned int u32x4;
typedef __attribute__((ext_vector_type(8))) int i32x8;
typedef __attribute__((ext_vector_type(4))) int i32x4;

#define CB   32
#define CC   256
#define CHW  3136            // 56*56
#define NTOT 25690112LL      // 32*256*3136
#define NPIX 100352          // 32*3136

__device__ __forceinline__ unsigned hash32(unsigned x) {
    x ^= x >> 16; x *= 0x7feb352du;
    x ^= x >> 15; x *= 0x846ca68bu;
    x ^= x >> 16; return x;
}

__device__ __forceinline__ float sigmoidf(float v) {
    return 1.0f / (1.0f + __expf(-v));
}

// ---------------------------------------------------------------------------
// Tensor Data Mover: DMA a 2D panel (rows x 64 f32 columns, row stride CHW
// elements) from global memory into LDS at byte offset lds_byte.
// Descriptor packing per CDNA5 ISA ch.8 (D# groups 0/1; 2D -> groups 2/3 = 0).
// This toolchain exposes the 6-arg builtin (4 groups + extra group + cpol).
// ---------------------------------------------------------------------------
__device__ __forceinline__ void tdm_load_panel(const float* gsrc,
                                               unsigned lds_byte, int rows) {
    unsigned long long ga = (unsigned long long)(const void*)gsrc;
    u32x4 g0;
    g0[0] = 1u;                                          // count=1 (valid), user
    g0[1] = lds_byte;                                    // lds_addr
    g0[2] = (unsigned)(ga & 0xffffffffu);                // global_addr[31:0]
    g0[3] = (unsigned)((ga >> 32) & 0x1ffffffu)          // global_addr[56:32]
          | (2u << 30);                                  // type = 2 ("image")
    const unsigned tdim0 = 64u, tile0 = 64u;
    const unsigned tdim1 = (unsigned)rows, tile1 = (unsigned)rows;
    const unsigned stride0 = (unsigned)CHW;              // dim0 stride (elems)
    i32x8 g1;
    g1[0] = (int)(2u << 16);                             // data_size = 4 bytes
    g1[1] = (int)((tdim0 & 0xffffu) << 16);              // tensor_dim0[15:0]
    g1[2] = (int)(((tdim0 >> 16) & 0xffffu)              // tensor_dim0[31:16]
          | ((tdim1 & 0xffffu) << 16));                  // tensor_dim1[15:0]
    g1[3] = (int)(((tdim1 >> 16) & 0xffffu)              // tensor_dim1[31:16]
          | (tile0 << 16));                              // tile_dim0
    g1[4] = (int)(tile1 & 0xffffu);                      // tile_dim1, tile_dim2=0
    g1[5] = (int)stride0;                                // tensor_dim0_stride lo
    g1[6] = 0;                                           // stride0 hi, stride1 lo
    g1[7] = 0;                                           // stride1 hi
    i32x4 gz4 = {0, 0, 0, 0};
    i32x8 gz8 = {0, 0, 0, 0, 0, 0, 0, 0};
    __builtin_amdgcn_tensor_load_to_lds(g0, g1, gz4, gz4, gz8, 0);
}

// ---------------------------------------------------------------------------
// K0: one block per (b,c). Applies per-chunk packet-loss mask (hash-based
// stand-in for the host-side permutation) while writing masked x to ws and
// reducing the (h,w) plane mean into y[b*C+c].
// ---------------------------------------------------------------------------
__global__ __launch_bounds__(256)
void k_mask_reduce(const float* __restrict__ x, float* __restrict__ xm,
                   float* __restrict__ y, const int* __restrict__ plr) {
    const int bc = blockIdx.x;            // 0..8191
    const int b  = bc >> 8;
    const int c  = bc & 255;
    const int t  = threadIdx.x;
    const unsigned rate = (unsigned)plr[0];
    const long long base = (long long)bc * CHW;   // x / xm are BCHW

    float acc = 0.0f;
    for (int hw = t; hw < CHW; hw += 256) {
        long long idx_bhwc = ((long long)b * CHW + hw) * CC + c;
        unsigned chunk = (unsigned)(idx_bhwc / 368);   // 1472B / 4B per chunk
        float keep = ((hash32(chunk ^ 0x9E3779B9u) % 100u) >= rate) ? 1.0f : 0.0f;
        float v = x[base + hw] * keep;
        xm[base + hw] = v;
        acc += v;
    }
    __shared__ float red[256];
    red[t] = acc;
    __syncthreads();
    for (int s = 128; s > 0; s >>= 1) {
        if (t < s) red[t] += red[t + s];
        __syncthreads();
    }
    if (t == 0) y[bc] = red[0] * (1.0f / (float)CHW);
}

// ---------------------------------------------------------------------------
// K0b: one block per batch. hidden = relu(y @ fc1^T), scores =
// sigmoid(hidden @ fc2^T), mask = relu(scores + rate*adapt - thr).
// ---------------------------------------------------------------------------
__global__ __launch_bounds__(256)
void k_se_mask(const float* __restrict__ y, const float* __restrict__ fc1,
               const float* __restrict__ fc2, const float* __restrict__ thr,
               const float* __restrict__ adapt, const int* __restrict__ plr,
               float* __restrict__ mask) {
    const int b = blockIdx.x;
    const int t = threadIdx.x;
    __shared__ float sy[CC];
    __shared__ float sh[16];
    sy[t] = y[b * CC + t];
    __syncthreads();
    if (t < 16) {
        float a = 0.0f;
        for (int c = 0; c < CC; ++c) a += fc1[t * CC + c] * sy[c];
        sh[t] = fmaxf(a, 0.0f);
    }
    __syncthreads();
    float a = 0.0f;
    for (int j = 0; j < 16; ++j) a += fc2[t * 16 + j] * sh[j];
    float score = sigmoidf(a);
    float m = score + (float)plr[0] * adapt[t] - thr[0];
    mask[b * CC + t] = fmaxf(m, 0.0f);
}

// ---------------------------------------------------------------------------
// WMMA fp32 GEMM over the pixel dimension, B staged via TDM.
//   A  : weights, row-major [256 x K] (L2-resident, streamed per wave)
//   B  : activations in BCHW layout; element (k, pixel p=b*3136+q) at
//        b*C*HW + k*HW + q.  EPI==2 runs two phases (xm panel, then xs panel).
//   Block: 256 threads = 8 waves; tile M=64 x N=64; wave = 16x32 subtile.
//   3136 % 64 == 0 so an N-tile never straddles a batch.
// EPI: 1 = xs = sigmoid(d)*aux   2 = relu(d)   3 = d * mask[b, m]
// ---------------------------------------------------------------------------
template <int EPI>
__global__ __launch_bounds__(256)
void k_gemm(const float* __restrict__ A, const float* __restrict__ B0,
            const float* __restrict__ B1, const float* __restrict__ aux,
            float* __restrict__ Dst, int K) {
    __shared__ float tile[CC * 64];      // 64KB B-panel (256 K-rows x 64 cols)

    const int lane = threadIdx.x & 31;
    const int w    = threadIdx.x >> 5;
    const int ll   = lane & 15;          // low lane group
    const int lh   = lane >> 4;          // 0 or 1 (half-wave)

    const int m0 = blockIdx.x * 64 + 16 * (w & 3);
    const long long n0blk = (long long)blockIdx.y * 64;
    const int b  = (int)(n0blk / CHW);
    const int q0 = (int)(n0blk % CHW);
    const long long colbase = (long long)b * CC * CHW + q0;
    const int jn0 = 32 * (w >> 2) + ll;  // local column of accumulator c0

    const unsigned lds0 = (unsigned)(unsigned long long)(const void*)&tile[0];

    v8f c0 = {};
    v8f c1 = {};

    const int nphase = (EPI == 2) ? 2 : 1;
    for (int ph = 0; ph < nphase; ++ph) {
        if (w == 0) {                    // one TDM issue per block (wave 0)
            const float* src = (ph == 0) ? B0 : B1;
            tdm_load_panel(src + colbase, lds0, CC);
            __builtin_amdgcn_s_wait_tensorcnt(0);
        }
        __syncthreads();                 // panel visible to all waves

        for (int k = 0; k < CC; k += 4) {
            const int kg = ph * CC + k;  // global K index into A
            // A fragment: lane ll -> row m0+ll; vgpr j -> K = kg + j + 2*lh
            const float* arow = A + (long long)(m0 + ll) * K + (kg + 2 * lh);
            __builtin_prefetch(arow + 16, 0, 1);
            v2f a; a[0] = arow[0]; a[1] = arow[1];

            // B fragment from LDS: row kb = k + 2*lh, col jn0 (and +16)
            const int kb = (k + 2 * lh) * 64;
            v2f bv0; bv0[0] = tile[kb + jn0];      bv0[1] = tile[kb + 64 + jn0];
            v2f bv1; bv1[0] = tile[kb + jn0 + 16]; bv1[1] = tile[kb + 64 + jn0 + 16];

            c0 = __builtin_amdgcn_wmma_f32_16x16x4_f32(false, a, false, bv0,
                                                       (short)0, c0, false, false);
            c1 = __builtin_amdgcn_wmma_f32_16x16x4_f32(false, a, false, bv1,
                                                       (short)0, c1, false, false);
        }
        if (ph + 1 < nphase) __syncthreads();   // all reads done before restage
    }

    // D layout: vgpr v, lane -> row m0 + v + 8*lh, col n0 + ll
    for (int v = 0; v < 8; ++v) {
        const int m = m0 + v + 8 * lh;
        const long long rowa = colbase + (long long)m * CHW;
        float d0 = c0[v];
        float d1 = c1[v];
        if (EPI == 1) {
            d0 = sigmoidf(d0) * aux[rowa + jn0];
            d1 = sigmoidf(d1) * aux[rowa + jn0 + 16];
        } else if (EPI == 2) {
            d0 = fmaxf(d0, 0.0f);
            d1 = fmaxf(d1, 0.0f);
        } else if (EPI == 3) {
            const float mk = aux[b * CC + m];
            d0 *= mk;
            d1 *= mk;
        }
        Dst[rowa + jn0]      = d0;
        Dst[rowa + jn0 + 16] = d1;
    }
}

// ---------------------------------------------------------------------------
extern "C" void kernel_launch(void* const* d_in, const int* in_sizes, int n_in,
                              void* d_out, int out_size, void* d_ws, size_t ws_size,
                              hipStream_t stream) {
    const float* x        = (const float*)d_in[0];
    const float* fc1_w    = (const float*)d_in[1];
    const float* fc2_w    = (const float*)d_in[2];
    const float* thr      = (const float*)d_in[3];
    const float* detect_w = (const float*)d_in[4];
    const float* rec1_w   = (const float*)d_in[5];
    const float* rec2_w   = (const float*)d_in[6];
    const float* adapt_w  = (const float*)d_in[7];
    const int*   plr      = (const int*)d_in[8];
    float* out = (float*)d_out;

    float* xm   = (float*)d_ws;          // masked x         (103 MB)
    float* xs   = xm + NTOT;             // sigmoid(det)*x   (103 MB)
    float* r1   = xs + NTOT;             // relu(rec1 @ z)   (103 MB)
    float* y    = r1 + NTOT;             // channel means    (32 KB)
    float* mask = y + CB * CC;           // gating mask      (32 KB)

    // K0: packet-loss mask + per-(b,c) mean
    k_mask_reduce<<<dim3(CB * CC), dim3(256), 0, stream>>>(x, xm, y, plr);
    // K0b: SE MLP -> mask
    k_se_mask<<<dim3(CB), dim3(256), 0, stream>>>(y, fc1_w, fc2_w, thr,
                                                  adapt_w, plr, mask);

    const dim3 gg(CC / 64, NPIX / 64);   // 4 x 1568 blocks
    // G1: xs = sigmoid(detect_w @ xm) * xm
    k_gemm<1><<<gg, dim3(256), 0, stream>>>(detect_w, xm, nullptr, xm, xs, CC);
    // G2: r1 = relu(rec1_w @ [xm; xs])
    k_gemm<2><<<gg, dim3(256), 0, stream>>>(rec1_w, xm, xs, nullptr, r1, 2 * CC);
    // G3: out = (rec2_w @ r1) * mask
    k_gemm<3><<<gg, dim3(256), 0, stream>>>(rec2_w, r1, nullptr, mask, out, CC);
}